// SSIMLoss_72232759984834
// MI455X (gfx1250) — compile-verified
//
#include <hip/hip_runtime.h>
#include <math.h>

typedef float v2f __attribute__((ext_vector_type(2)));
typedef float v8f __attribute__((ext_vector_type(8)));

#define WAVES_PER_BLOCK 4
#define HSTRIDE 17                 // padded LDS row stride (floats) to stagger banks
#define HROWS 32
#define HQ_SIZE (HROWS * HSTRIDE)  // 544 floats per blurred quantity
#define IMG 512
#define OUT 502
#define NIMG 32
#define TILES_1D 32                            // ceil(502/16)
#define TOTAL_TILES (NIMG * TILES_1D * TILES_1D)  // 32768 waves
#define NBLOCKS (TOTAL_TILES / WAVES_PER_BLOCK)   // 8192 blocks
#define N_ELEMS 8064128.0f                     // 32 * 502 * 502

// D(16x16 f32) = A(16x4 f32) x B(4x16 f32) + C : full-precision WMMA
static __device__ __forceinline__ v8f wmma4(v2f a, v2f b, v8f c) {
    return __builtin_amdgcn_wmma_f32_16x16x4_f32(
        /*neg_a=*/false, a, /*neg_b=*/false, b,
        /*c_mod=*/(short)0, c, /*reuse_a=*/false, /*reuse_b=*/false);
}

__global__ void __launch_bounds__(128)
ssim_tile_kernel(const float* __restrict__ P, const float* __restrict__ G,
                 float* __restrict__ partial) {
    __shared__ float gw[48];                          // zero-padded gaussian LUT, g at [16..26]
    __shared__ float Hbuf[WAVES_PER_BLOCK][5 * HQ_SIZE];
    __shared__ float redbuf[WAVES_PER_BLOCK];

    const int tid = threadIdx.x;

    // Build the normalized 11-tap gaussian window (matches reference: float64 math -> f32)
    if (tid < 48) gw[tid] = 0.0f;
    __syncthreads();
    if (tid < 11) {
        double s = 0.0;
        for (int t = 0; t < 11; ++t) {
            double c = (double)(t - 5);
            s += exp(-c * c / 4.5);
        }
        double c = (double)(tid - 5);
        gw[16 + tid] = (float)(exp(-c * c / 4.5) / s);
    }
    __syncthreads();

    const int wv   = tid >> 5;
    const int lane = tid & 31;
    const int n    = lane & 15;    // column (stage1 B / stage2 A row index)
    const int hi   = lane >> 4;    // half-wave select
    const int s0   = hi * 2;       // K-subelement base: lanes<16 -> {0,1}, lanes>=16 -> {2,3}

    // Per-lane constant gaussian operand chunks; identical register image serves as
    // stage-1 B-matrix (G[m,j] = g[m-j]) and stage-2 A-matrix (Gv[r,m] = g[m-r]).
    v2f gk[7];
#pragma unroll
    for (int kk = 0; kk < 7; ++kk) {
        int m0 = 4 * kk + s0;                 // m0 in [0,27], index 16+m0-n in [1,43]
        v2f t;
        t.x = gw[16 + m0 - n];
        t.y = gw[16 + m0 + 1 - n];
        gk[kk] = t;
    }

    const int waveId = blockIdx.x * WAVES_PER_BLOCK + wv;
    const int b  = waveId >> 10;
    const int tt = waveId & 1023;
    const int r0 = (tt >> 5) << 4;
    const int c0 = (tt & 31) << 4;

    const float* Pimg = P + (size_t)b * (IMG * IMG);
    const float* Gimg = G + (size_t)b * (IMG * IMG);
    float* H = Hbuf[wv];

    // ---- Stage 1: horizontal gaussian of {p, g, p*p, g*g, p*g} via WMMA ----
    // H rows 0..25 are valid; rows 26..31 hold clamped garbage killed by zero weights.
#pragma unroll
    for (int h = 0; h < 2; ++h) {
        v8f a0 = {0,0,0,0,0,0,0,0}, a1 = a0, a2 = a0, a3 = a0, a4 = a0;
        int grow = r0 + h * 16 + n;            // A-matrix row = lane&15
        if (grow > IMG - 1) grow = IMG - 1;    // clamp (edge tiles; masked later)
        const float* Prow = Pimg + grow * IMG;
        const float* Grow = Gimg + grow * IMG;
#pragma unroll
        for (int kk = 0; kk < 7; ++kk) {
            int ca = c0 + 4 * kk + s0;
            int cb = ca + 1;
            if (ca > IMG - 1) ca = IMG - 1;
            if (cb > IMG - 1) cb = IMG - 1;
            float p0 = Prow[ca], p1 = Prow[cb];
            float q0 = Grow[ca], q1 = Grow[cb];
            v2f Ap  = {p0, p1};
            v2f Ag  = {q0, q1};
            v2f App = {p0 * p0, p1 * p1};
            v2f Agg = {q0 * q0, q1 * q1};
            v2f Apg = {p0 * q0, p1 * q1};
            a0 = wmma4(Ap,  gk[kk], a0);
            a1 = wmma4(Ag,  gk[kk], a1);
            a2 = wmma4(App, gk[kk], a2);
            a3 = wmma4(Agg, gk[kk], a3);
            a4 = wmma4(Apg, gk[kk], a4);
        }
        // C/D layout: lane<16 holds rows v, lane>=16 holds rows v+8, col = n
        int rb = h * 16 + hi * 8;
#pragma unroll
        for (int v = 0; v < 8; ++v) {
            int off = (rb + v) * HSTRIDE + n;
            H[0 * HQ_SIZE + off] = a0[v];
            H[1 * HQ_SIZE + off] = a1[v];
            H[2 * HQ_SIZE + off] = a2[v];
            H[3 * HQ_SIZE + off] = a3[v];
            H[4 * HQ_SIZE + off] = a4[v];
        }
    }
    // Same-wave LDS producer/consumer: DS ops in-order per wave; compiler waits dscnt.

    // ---- Stage 2: vertical gaussian via WMMA (A = constant gk, B from LDS) ----
    v8f m1 = {0,0,0,0,0,0,0,0}, m2 = m1, e11 = m1, e22 = m1, e12 = m1;
#pragma unroll
    for (int kk = 0; kk < 7; ++kk) {
        int m0 = 4 * kk + s0;
        int o0 = m0 * HSTRIDE + n;
        int o1 = o0 + HSTRIDE;
        v2f B0 = {H[0 * HQ_SIZE + o0], H[0 * HQ_SIZE + o1]};
        v2f B1 = {H[1 * HQ_SIZE + o0], H[1 * HQ_SIZE + o1]};
        v2f B2 = {H[2 * HQ_SIZE + o0], H[2 * HQ_SIZE + o1]};
        v2f B3 = {H[3 * HQ_SIZE + o0], H[3 * HQ_SIZE + o1]};
        v2f B4 = {H[4 * HQ_SIZE + o0], H[4 * HQ_SIZE + o1]};
        m1  = wmma4(gk[kk], B0, m1);
        m2  = wmma4(gk[kk], B1, m2);
        e11 = wmma4(gk[kk], B2, e11);
        e22 = wmma4(gk[kk], B3, e22);
        e12 = wmma4(gk[kk], B4, e12);
    }

    // ---- SSIM map + masked accumulate ----
    const float C1 = 0.009801f;   // (0.01 * 9.9)^2
    const float C2 = 0.088209f;   // (0.03 * 9.9)^2
    float acc = 0.0f;
#pragma unroll
    for (int v = 0; v < 8; ++v) {
        int row = r0 + v + hi * 8;
        int col = c0 + n;
        float mu1 = m1[v], mu2 = m2[v];
        float mu1sq = mu1 * mu1, mu2sq = mu2 * mu2, mu12 = mu1 * mu2;
        float s1  = e11[v] - mu1sq;
        float s2  = e22[v] - mu2sq;
        float s12 = e12[v] - mu12;
        float cs  = (2.0f * s12 + C2) / (s1 + s2 + C2);
        float ss  = ((2.0f * mu12 + C1) / (mu1sq + mu2sq + C1)) * cs;
        if (row < OUT && col < OUT) acc += ss;
    }
    // wave32 reduce
#pragma unroll
    for (int o = 16; o > 0; o >>= 1) acc += __shfl_xor(acc, o, 32);
    if (lane == 0) redbuf[wv] = acc;
    __syncthreads();
    if (tid == 0)
        partial[blockIdx.x] = redbuf[0] + redbuf[1] + redbuf[2] + redbuf[3];
}

__global__ void __launch_bounds__(256)
ssim_reduce_kernel(const float* __restrict__ partial, float* __restrict__ out) {
    __shared__ float s[256];
    float a = 0.0f;
    for (int i = threadIdx.x; i < NBLOCKS; i += 256) a += partial[i];
    s[threadIdx.x] = a;
    __syncthreads();
    for (int w = 128; w > 0; w >>= 1) {
        if ((int)threadIdx.x < w) s[threadIdx.x] += s[threadIdx.x + w];
        __syncthreads();
    }
    if (threadIdx.x == 0) out[0] = 1.0f - s[0] / N_ELEMS;
}

extern "C" void kernel_launch(void* const* d_in, const int* in_sizes, int n_in,
                              void* d_out, int out_size, void* d_ws, size_t ws_size,
                              hipStream_t stream) {
    (void)in_sizes; (void)n_in; (void)out_size; (void)ws_size;
    const float* P = (const float*)d_in[0];   // predict, f32 (32,1,512,512)
    const float* G = (const float*)d_in[1];   // gt,      f32 (32,1,512,512)
    float* partial = (float*)d_ws;            // 8192 floats of scratch

    ssim_tile_kernel<<<NBLOCKS, 128, 0, stream>>>(P, G, partial);
    ssim_reduce_kernel<<<1, 256, 0, stream>>>(partial, (float*)d_out);
}